// InvConditionalLinear_17961553232465
// MI455X (gfx1250) — compile-verified
//
#include <hip/hip_runtime.h>
#include <hip/hip_bf16.h>

// MI455X / gfx1250: wave32, WMMA FP32 16x16x4.
// One wave32 per pixel: lane = matrix column, register = matrix row.
// 2x2-blocked LU (16x16 blocks), Schur update D - L21*U12 on v_wmma_f32_16x16x4_f32.

typedef __attribute__((ext_vector_type(2))) float v2f;
typedef __attribute__((ext_vector_type(8))) float v8f;

#define BATCH 16
#define C32   32
#define HW    2304            // 48*48
#define NPIX  (BATCH * HW)    // 36864
#define ZSIZE (BATCH * C32 * HW)

// uniform broadcast of lane `l` (l is a compile-time constant after unroll -> v_readlane)
__device__ __forceinline__ float bcast(float v, int l) {
  return __builtin_bit_cast(float, __builtin_amdgcn_readlane(__builtin_bit_cast(int, v), l));
}
// cross-half swap (lanes 0-15 <-> 16-31)
__device__ __forceinline__ float xor16(float v) {
  return __shfl_xor(v, 16, 32);
}

__global__ __launch_bounds__(256) void invcl_solve_kernel(
    const float* __restrict__ xin,   // [B, 32, 48, 48]
    const float* __restrict__ wgt,   // [B, 1024, 48, 48]
    float* __restrict__ zout,        // [B, 32, 48, 48]
    float* __restrict__ ws)          // [NPIX] per-pixel log|det|
{
  const int lane = threadIdx.x & 31;
  const int wid  = threadIdx.x >> 5;
  const int P    = blockIdx.x * 8 + wid;     // pixel id, 0..36863 (exact grid)
  const int b    = P / HW;
  const int p    = P - b * HW;
  const bool lo  = lane < 16;

  // ---- load W: a[i] = W[row i][col lane] ----
  const float* wp = wgt + (size_t)b * (1024u * HW) + p;
  float a[32];
#pragma unroll
  for (int i = 0; i < 32; ++i)
    a[i] = wp[(size_t)(i * 32 + lane) * HW];

  // ---- load rhs replicated across lanes (wave-uniform addresses) ----
  const float* xp = xin + (size_t)b * (32u * HW) + p;
  float bv[32];
#pragma unroll
  for (int i = 0; i < 32; ++i)
    bv[i] = xp[(size_t)i * HW];

  // L21 (negated multipliers) accumulated directly in WMMA A-matrix layout:
  // chunk c (K=4c..4c+3): la[2c+ (K&1)], lanes 0-15 = K even-half, 16-31 = K+2 half
  float la[8];
#pragma unroll
  for (int r = 0; r < 8; ++r) la[r] = 0.0f;

  float ld = 0.0f;  // sum of log2|pivot|

  // ================= panel factorization: k = 0..15 =================
#pragma unroll
  for (int k = 0; k < 16; ++k) {
    const float piv = bcast(a[k], k);
    ld += __log2f(__builtin_fabsf(piv));
    const float ip = __builtin_amdgcn_rcpf(piv);

    // rows k+1..15: full-width update (forms U11 and U12)
#pragma unroll
    for (int i = k + 1; i < 16; ++i) {
      const float nm = -bcast(a[i], k) * ip;
      a[i] += nm * a[k];
      bv[i] += nm * bv[k];
    }

    // rows 16..31: update panel columns only (defer D block -> WMMA Schur)
    const int Kc  = k & 3;
    const int ch  = k >> 2;
    const int rg  = 2 * ch + (Kc & 1);
    const int off = (Kc >> 1) * 16;
#pragma unroll
    for (int i = 16; i < 32; ++i) {
      const float nm  = -bcast(a[i], k) * ip;
      const float nmm = lo ? nm : 0.0f;       // mask off the D block
      a[i] += nmm * a[k];
      bv[i] += nm * bv[k];                    // rhs forward-eliminated in full
      la[rg] = (lane == (i - 16) + off) ? nm : la[rg];   // deposit -m into A layout
    }
  }

  // ================= Schur complement: S = (-L21)*U12 + D via WMMA ==========
  // Pack D (a[16..31], lanes 16-31) into C/D layout.
  v8f cacc;
#pragma unroll
  for (int r = 0; r < 8; ++r) {
    const float t = xor16(a[16 + r]);         // row r of D into lanes 0-15
    cacc[r] = lo ? t : a[24 + r];             // row r+8 already in lanes 16-31
  }
  // 4 x v_wmma_f32_16x16x4_f32 accumulate K = 0..15
#pragma unroll
  for (int c = 0; c < 4; ++c) {
    v2f Am;
    Am[0] = la[2 * c];
    Am[1] = la[2 * c + 1];
    v2f Bm;                                    // U12 rows 4c..4c+3 -> B layout
    const float t0 = xor16(a[4 * c]);
    const float t1 = xor16(a[4 * c + 1]);
    Bm[0] = lo ? t0 : a[4 * c + 2];
    Bm[1] = lo ? t1 : a[4 * c + 3];
    cacc = __builtin_amdgcn_wmma_f32_16x16x4_f32(
        false, Am, false, Bm, (short)0, cacc, false, false);
  }
  // Unpack S back into elimination layout (lanes 16-31 of a[16..31])
#pragma unroll
  for (int r = 0; r < 8; ++r) {
    const float t = xor16(cacc[r]);
    a[16 + r] = lo ? a[16 + r] : t;
    a[24 + r] = lo ? a[24 + r] : cacc[r];
  }

  // ================= trailing factorization: k = 16..31 =================
#pragma unroll
  for (int k = 16; k < 32; ++k) {
    const float piv = bcast(a[k], k);
    ld += __log2f(__builtin_fabsf(piv));
    const float ip = __builtin_amdgcn_rcpf(piv);
#pragma unroll
    for (int i = k + 1; i < 32; ++i) {
      const float nm = -bcast(a[i], k) * ip;
      a[i] += nm * a[k];
      bv[i] += nm * bv[k];
    }
  }

  // ================= back substitution =================
  float zz = 0.0f;
#pragma unroll
  for (int k = 31; k >= 0; --k) {
    const float ukk = bcast(a[k], k);
    const float zk  = bv[k] * __builtin_amdgcn_rcpf(ukk);
    zz = (lane == k) ? zk : zz;               // lane k keeps z[k]
#pragma unroll
    for (int i = 0; i < k; ++i)
      bv[i] -= zk * bcast(a[i], k);
  }

  zout[(size_t)(b * 32 + lane) * HW + p] = zz;
  if (lane == 0)
    ws[P] = ld * 0.6931471805599453f;          // log2 -> ln
}

// out_logdet[b] = logdet_in[b] - sum over 2304 pixels of ws
__global__ __launch_bounds__(256) void invcl_reduce_kernel(
    const float* __restrict__ ws,
    const float* __restrict__ logdet_in,
    float* __restrict__ out_logdet)
{
  const int b = blockIdx.x;
  float s = 0.0f;
  for (int i = threadIdx.x; i < HW; i += 256)
    s += ws[b * HW + i];
#pragma unroll
  for (int off = 16; off > 0; off >>= 1)
    s += __shfl_down(s, off, 32);
  __shared__ float sm[8];
  if ((threadIdx.x & 31) == 0) sm[threadIdx.x >> 5] = s;
  __syncthreads();
  if (threadIdx.x == 0) {
    float t = 0.0f;
#pragma unroll
    for (int w = 0; w < 8; ++w) t += sm[w];
    out_logdet[b] = logdet_in[b] - t;
  }
}

extern "C" void kernel_launch(void* const* d_in, const int* in_sizes, int n_in,
                              void* d_out, int out_size, void* d_ws, size_t ws_size,
                              hipStream_t stream) {
  const float* inp    = (const float*)d_in[0];  // [16,32,48,48]
  const float* wgt    = (const float*)d_in[1];  // [16,1024,48,48]
  const float* logdet = (const float*)d_in[2];  // [16]
  float* out  = (float*)d_out;
  float* zs   = out;                 // first ZSIZE floats
  float* ld2  = out + ZSIZE;         // last 16 floats
  float* ws   = (float*)d_ws;        // NPIX floats of scratch

  invcl_solve_kernel<<<NPIX / 8, 256, 0, stream>>>(inp, wgt, zs, ws);
  invcl_reduce_kernel<<<BATCH, 256, 0, stream>>>(ws, logdet, ld2);
}